// GNNLayer_30425548324951
// MI455X (gfx1250) — compile-verified
//
#include <hip/hip_runtime.h>

typedef float v2f __attribute__((ext_vector_type(2)));
typedef float v8f __attribute__((ext_vector_type(8)));
typedef unsigned int uint32x4 __attribute__((ext_vector_type(4)));
typedef int int32x8 __attribute__((ext_vector_type(8)));
typedef int int32x4 __attribute__((ext_vector_type(4)));

#define Dd 128
#define NN 200
#define BB 8
#define MH (BB*NN)          // 1600 h-rows
#define NT 13               // ceil(200/16) j-tiles
#define LSTR 132            // LDS row stride (floats): 128 data + 4 pad (TDM pad)

// ---------------------------------------------------------------------------
// K1: four h-side GEMMs  O = h @ W.T + b   for W in {Uw,Vw,Aw,Bw}
// grid 100 blocks (16 rows each), 128 threads = 4 waves, wave w -> matrix w
// ---------------------------------------------------------------------------
__global__ __launch_bounds__(128)
void k1_h_gemms(const float* __restrict__ h,
                const float* __restrict__ Uw, const float* __restrict__ Ub,
                const float* __restrict__ Vw, const float* __restrict__ Vb,
                const float* __restrict__ Aw, const float* __restrict__ Ab,
                const float* __restrict__ Bw, const float* __restrict__ Bb,
                float* __restrict__ Uh, float* __restrict__ Vh,
                float* __restrict__ Ah, float* __restrict__ Bh)
{
    __shared__ float lh[16 * LSTR];
    const int t    = threadIdx.x;
    const int wave = t >> 5;          // 0..3
    const int lane = t & 31;
    const int hl   = lane >> 4;       // half-wave: 0 or 1
    const int l15  = lane & 15;
    const int mt   = blockIdx.x;      // 0..99

    for (int idx = t; idx < 16 * Dd; idx += 128) {
        int r = idx >> 7, c = idx & 127;
        lh[r * LSTR + c] = h[(mt * 16 + r) * Dd + c];
    }
    __syncthreads();

    const float* Wt[4]   = {Uw, Vw, Aw, Bw};
    const float* biasT[4]= {Ub, Vb, Ab, Bb};
    float*       Ot[4]   = {Uh, Vh, Ah, Bh};
    const float* W    = Wt[wave];
    const float* bias = biasT[wave];
    float*       O    = Ot[wave];

    for (int nt = 0; nt < 8; ++nt) {
        const int n = nt * 16 + l15;
        v8f acc = {};
        #pragma unroll
        for (int ks = 0; ks < 32; ++ks) {
            const int kk = ks * 4 + hl * 2;
            v2f a; a.x = lh[l15 * LSTR + kk]; a.y = lh[l15 * LSTR + kk + 1];
            v2f b; b.x = W[n * Dd + kk];      b.y = W[n * Dd + kk + 1];
            acc = __builtin_amdgcn_wmma_f32_16x16x4_f32(
                    false, a, false, b, (short)0, acc, false, false);
        }
        const float bv = bias[n];
        #pragma unroll
        for (int v = 0; v < 8; ++v) {
            const int m = mt * 16 + v + hl * 8;
            O[m * Dd + n] = acc[v] + bv;
        }
    }
}

// ---------------------------------------------------------------------------
// TDM issue: load one [rows x 128 f32] tile (rows<=16, zero-filled OOB) into
// LDS with 4-DWORD padding every 128 DWORDs (row stride 132 floats).
// D# packing per CDNA5 ISA 8.3/8.4. 2D tensor -> groups 2/3 zero.
// This toolchain uses the 6-arg builtin: (v4u, v8i, v4i, v4i, v8i, i32 cpol);
// the trailing v8i is zero-filled (unused for a single 2D descriptor).
// ---------------------------------------------------------------------------
__device__ __forceinline__
void tdm_load_tile(const float* gsrc, unsigned lds_byte_addr, int rows_remaining)
{
    unsigned long long ga = (unsigned long long)(const void*)gsrc;

    uint32x4 g0;
    g0[0] = 1u;                                   // count=1, user mode
    g0[1] = lds_byte_addr;                        // lds_addr
    g0[2] = (unsigned)(ga & 0xFFFFFFFFu);         // global_addr[31:0]
    g0[3] = (unsigned)((ga >> 32) & 0x1FFFFFFu)   // global_addr[56:32]
          | (2u << 30);                           // type = 2 (image)

    int32x8 g1;
    // data_size=2 (4B) | pad_enable | pad_interval=6 (128 DW) | pad_amount=3 (4 DW)
    g1[0] = (int)((2u << 16) | (1u << 20) | (6u << 22) | (3u << 25));
    g1[1] = (int)(128u << 16);                    // tensor_dim0 = 128 (bits 79:48)
    g1[2] = (int)((unsigned)rows_remaining << 16);// tensor_dim1 (bits 111:80)
    g1[3] = (int)(128u << 16);                    // tile_dim0 = 128 (bits 127:112)
    g1[4] = 16;                                   // tile_dim1 = 16, tile_dim2 = 0
    g1[5] = 128;                                  // tensor_dim0_stride = 128
    g1[6] = 0;                                    // stride0 hi / dim1_stride lo
    g1[7] = 0;

    int32x4 g2 = {};                              // 2D: dims 2/3 unused
    int32x4 g3 = {};
    int32x8 g4 = {};                              // trailing group: unused/zero

    __builtin_amdgcn_tensor_load_to_lds(g0, g1, g2, g3, g4, 0);
}

// ---------------------------------------------------------------------------
// K2: main e pass. Block = (b,i). e_new = e@Cw.T + Cb + Ah[b,j,:] + Bh[b,i,:]
//     -> store e_new into d_out e-region, accumulate gated agg + BN partials.
// 8 waves, wave w owns channel tile d = [16w, 16w+16). TDM double-buffers the
// 16x128 e tile into LDS; wave 0 drives the DMA, s_wait_tensorcnt fences.
// ---------------------------------------------------------------------------
__global__ __launch_bounds__(256)
void k2_e_pass(const float* __restrict__ e,
               const float* __restrict__ Cw, const float* __restrict__ Cb,
               const float* __restrict__ Vh, const float* __restrict__ Ah,
               const float* __restrict__ Bh,
               float* __restrict__ e_new_out,     // [1600,200,128] (d_out e region)
               float* __restrict__ agg,           // [1600,128]
               float* __restrict__ epart_sum,     // [1600,128]
               float* __restrict__ epart_sq)      // [1600,128]
{
    __shared__ float le[2][16 * LSTR];
    const int t    = threadIdx.x;
    const int wave = t >> 5;          // 0..7 == N-tile
    const int lane = t & 31;
    const int hl   = lane >> 4;
    const int l15  = lane & 15;
    const int blk  = blockIdx.x;      // 0..1599
    const int b    = blk / NN;
    const int d    = wave * 16 + l15; // channel

    const unsigned lds0 = (unsigned)(size_t)(void*)&le[0][0];
    const unsigned lds1 = (unsigned)(size_t)(void*)&le[1][0];

    // Preload Cw B-fragments for this wave's N-tile: kept in 64 VGPRs.
    v2f bf[32];
    #pragma unroll
    for (int ks = 0; ks < 32; ++ks) {
        const int kk = ks * 4 + hl * 2;
        bf[ks].x = Cw[d * Dd + kk];
        bf[ks].y = Cw[d * Dd + kk + 1];
    }
    const float bi_d = Bh[blk * Dd + d];
    const float cb_d = Cb[d];

    float aggp = 0.f, sump = 0.f, sqp = 0.f;
    const float* erow = e + (size_t)blk * NN * Dd;
    float* eno = e_new_out + (size_t)blk * NN * Dd;

    if (wave == 0)
        tdm_load_tile(erow, lds0, NN);            // tile 0 -> buf0

    for (int jt = 0; jt < NT; ++jt) {
        if (wave == 0) {
            if (jt + 1 < NT) {
                tdm_load_tile(erow + (size_t)(jt + 1) * 16 * Dd,
                              ((jt + 1) & 1) ? lds1 : lds0,
                              NN - (jt + 1) * 16);
                __builtin_amdgcn_s_wait_tensorcnt(1);  // tile jt complete
            } else {
                __builtin_amdgcn_s_wait_tensorcnt(0);
            }
        }
        __syncthreads();

        const float* lt = &le[jt & 1][0];
        v8f acc = {};
        #pragma unroll
        for (int ks = 0; ks < 32; ++ks) {
            const int kk = ks * 4 + hl * 2;
            v2f a; a.x = lt[l15 * LSTR + kk]; a.y = lt[l15 * LSTR + kk + 1];
            acc = __builtin_amdgcn_wmma_f32_16x16x4_f32(
                    false, a, false, bf[ks], (short)0, acc, false, false);
        }

        #pragma unroll
        for (int v = 0; v < 8; ++v) {
            const int j = jt * 16 + v + hl * 8;
            if (j < NN) {
                float en = acc[v] + cb_d + bi_d + Ah[(b * NN + j) * Dd + d];
                eno[(size_t)j * Dd + d] = en;
                sump += en;
                sqp  += en * en;
                float g = 1.f / (1.f + __expf(-en));
                aggp += g * Vh[(b * NN + j) * Dd + d];
            }
        }
        __syncthreads();   // all waves done reading buf[jt&1] before reuse
    }

    // lane l and l^16 hold the same channel d with disjoint j's -> fold.
    aggp += __shfl_xor(aggp, 16, 32);
    sump += __shfl_xor(sump, 16, 32);
    sqp  += __shfl_xor(sqp, 16, 32);
    if (hl == 0) {
        agg[blk * Dd + d]       = aggp;
        epart_sum[blk * Dd + d] = sump;
        epart_sq[blk * Dd + d]  = sqp;
    }
}

// ---------------------------------------------------------------------------
// K3: h_new = Uh + agg, per-block BN partials. grid 100, 128 threads (one/ch)
// ---------------------------------------------------------------------------
__global__ __launch_bounds__(128)
void k3_hnew(const float* __restrict__ Uh, const float* __restrict__ agg,
             float* __restrict__ h_new,
             float* __restrict__ hpart_sum, float* __restrict__ hpart_sq)
{
    const int d   = threadIdx.x;
    const int blk = blockIdx.x;
    float s = 0.f, q = 0.f;
    for (int r = 0; r < 16; ++r) {
        int m = blk * 16 + r;
        float v = Uh[m * Dd + d] + agg[m * Dd + d];
        h_new[m * Dd + d] = v;
        s += v; q += v * v;
    }
    hpart_sum[blk * Dd + d] = s;
    hpart_sq[blk * Dd + d]  = q;
}

// ---------------------------------------------------------------------------
// K4: finalize BN stats (deterministic tree reduction). grid 128 (one/channel)
// stats layout: [0:128) e_mu, [128:256) e_rs, [256:384) h_mu, [384:512) h_rs
// ---------------------------------------------------------------------------
__global__ __launch_bounds__(256)
void k4_stats(const float* __restrict__ eps_, const float* __restrict__ eqs_,
              const float* __restrict__ hps_, const float* __restrict__ hqs_,
              float* __restrict__ stats)
{
    __shared__ float red[256];
    const int ch = blockIdx.x;
    const int t  = threadIdx.x;
    float es = 0.f, eq = 0.f, hs = 0.f, hq = 0.f;
    for (int p = t; p < MH; p += 256)  { es += eps_[p * Dd + ch]; eq += eqs_[p * Dd + ch]; }
    for (int p = t; p < 100; p += 256) { hs += hps_[p * Dd + ch]; hq += hqs_[p * Dd + ch]; }

    auto blockReduce = [&](float v) -> float {
        red[t] = v; __syncthreads();
        for (int s = 128; s > 0; s >>= 1) {
            if (t < s) red[t] += red[t + s];
            __syncthreads();
        }
        float r = red[0]; __syncthreads();
        return r;
    };
    float ES = blockReduce(es), EQ = blockReduce(eq);
    float HS = blockReduce(hs), HQ = blockReduce(hq);

    if (t == 0) {
        const float ce = 1.f / (float)(BB * NN * NN);
        float mu  = ES * ce;
        float var = EQ * ce - mu * mu;
        stats[ch]       = mu;
        stats[128 + ch] = rsqrtf(var + 1e-5f);
        const float chn = 1.f / (float)MH;
        float mh = HS * chn;
        float vh = HQ * chn - mh * mh;
        stats[256 + ch] = mh;
        stats[384 + ch] = rsqrtf(vh + 1e-5f);
    }
}

// ---------------------------------------------------------------------------
// K5: h_out = h_in + relu(bn(h_new)). grid 800 x 256
// ---------------------------------------------------------------------------
__global__ __launch_bounds__(256)
void k5_hout(const float* __restrict__ h_in, const float* __restrict__ h_new,
             const float* __restrict__ gamma, const float* __restrict__ beta,
             const float* __restrict__ stats, float* __restrict__ h_out)
{
    const int idx = blockIdx.x * 256 + threadIdx.x;   // < 204800
    const int d   = idx & 127;
    float v = (h_new[idx] - stats[256 + d]) * stats[384 + d] * gamma[d] + beta[d];
    h_out[idx] = h_in[idx] + fmaxf(v, 0.f);
}

// ---------------------------------------------------------------------------
// K6: e_out = e_in + relu(bn(e_new)), in-place over d_out e region (float4)
// grid 40000 x 256
// ---------------------------------------------------------------------------
__global__ __launch_bounds__(256)
void k6_eout(const float* __restrict__ e_in,
             const float* __restrict__ gamma, const float* __restrict__ beta,
             const float* __restrict__ stats, float* __restrict__ e_io)
{
    const size_t f = (size_t)blockIdx.x * 256 + threadIdx.x;  // float4 index
    const float4* ein4 = (const float4*)e_in;
    float4* eo4 = (float4*)e_io;
    const int d0 = (int)((f & 31) * 4);      // channel of component .x

    float4 en = eo4[f];     // e_new stashed by K2
    float4 ei = ein4[f];
    float4 r;
    r.x = ei.x + fmaxf((en.x - stats[d0+0]) * stats[128+d0+0] * gamma[d0+0] + beta[d0+0], 0.f);
    r.y = ei.y + fmaxf((en.y - stats[d0+1]) * stats[128+d0+1] * gamma[d0+1] + beta[d0+1], 0.f);
    r.z = ei.z + fmaxf((en.z - stats[d0+2]) * stats[128+d0+2] * gamma[d0+2] + beta[d0+2], 0.f);
    r.w = ei.w + fmaxf((en.w - stats[d0+3]) * stats[128+d0+3] * gamma[d0+3] + beta[d0+3], 0.f);
    eo4[f] = r;
}

// ---------------------------------------------------------------------------
extern "C" void kernel_launch(void* const* d_in, const int* in_sizes, int n_in,
                              void* d_out, int out_size, void* d_ws, size_t ws_size,
                              hipStream_t stream)
{
    const float* h       = (const float*)d_in[0];
    const float* e       = (const float*)d_in[1];
    const float* Uw      = (const float*)d_in[2];
    const float* Ub      = (const float*)d_in[3];
    const float* Vw      = (const float*)d_in[4];
    const float* Vb      = (const float*)d_in[5];
    const float* Aw      = (const float*)d_in[6];
    const float* Ab      = (const float*)d_in[7];
    const float* Bw      = (const float*)d_in[8];
    const float* Bb      = (const float*)d_in[9];
    const float* Cw      = (const float*)d_in[10];
    const float* Cb      = (const float*)d_in[11];
    const float* gamma_h = (const float*)d_in[12];
    const float* beta_h  = (const float*)d_in[13];
    const float* gamma_e = (const float*)d_in[14];
    const float* beta_e  = (const float*)d_in[15];

    float* out   = (float*)d_out;
    float* h_out = out;                  // 204800 floats
    float* e_out = out + MH * Dd;        // 40,960,000 floats (also e_new staging)

    float* ws       = (float*)d_ws;
    float* Uh       = ws;                  // 204800
    float* Vh       = ws + 204800;         // 204800
    float* Ah       = ws + 409600;         // 204800
    float* Bh       = ws + 614400;         // 204800
    float* agg      = ws + 819200;         // 204800
    float* h_new    = ws + 1024000;        // 204800
    float* eps_     = ws + 1228800;        // 204800  e partial sums
    float* eqs_     = ws + 1433600;        // 204800  e partial sumsq
    float* hps_     = ws + 1638400;        // 12800   h partial sums
    float* hqs_     = ws + 1651200;        // 12800   h partial sumsq
    float* stats    = ws + 1664000;        // 512

    k1_h_gemms<<<100, 128, 0, stream>>>(h, Uw, Ub, Vw, Vb, Aw, Ab, Bw, Bb,
                                        Uh, Vh, Ah, Bh);
    k2_e_pass<<<MH, 256, 0, stream>>>(e, Cw, Cb, Vh, Ah, Bh,
                                      e_out, agg, eps_, eqs_);
    k3_hnew<<<100, 128, 0, stream>>>(Uh, agg, h_new, hps_, hqs_);
    k4_stats<<<128, 256, 0, stream>>>(eps_, eqs_, hps_, hqs_, stats);
    k5_hout<<<800, 256, 0, stream>>>(h, h_new, gamma_h, beta_h, stats, h_out);
    k6_eout<<<40000, 256, 0, stream>>>(e, gamma_e, beta_e, stats, e_out);
}